// Gaussian_Distance_1580547971931
// MI455X (gfx1250) — compile-verified
//
#include <hip/hip_runtime.h>
#include <hip/hip_bf16.h>
#include <stdint.h>

typedef unsigned int u32;
typedef unsigned long long u64;
typedef unsigned int v4u __attribute__((ext_vector_type(4)));
typedef int          v8i __attribute__((ext_vector_type(8)));
typedef int          v4i __attribute__((ext_vector_type(4)));

#define NROWS 256            // concatenated a|b samples
#define D     4096           // 16*16*16 pooled features
#define KT    128            // LDS tile width (floats per row)
#define KCHUNK 512           // k-slab per block
#define KBLK  (D / KCHUNK)   // 8
#define NCHUNK (KCHUNK / KT) // 4 inner tiles per block
#define LOG2E 1.4426950408889634f
#define NEG_HALF_LOG2E (-0.72134752044448169f)

// ---------------------------------------------------------------------------
// Stage 1: 2x2 avg-pool of mu -> X rows, and avg-pool(exp(logvar))/4 -> V rows
// X,V are [256][4096] f32; rows 0..127 from (mu_a,logvar_a), 128..255 from b.
// ---------------------------------------------------------------------------
__global__ __launch_bounds__(256) void pool_kernel(
    const float* __restrict__ mu_a, const float* __restrict__ lv_a,
    const float* __restrict__ mu_b, const float* __restrict__ lv_b,
    float* __restrict__ X, float* __restrict__ V) {
  u32 idx = blockIdx.x * 256u + threadIdx.x;            // 0 .. 256*4096-1
  if (idx >= NROWS * D) return;
  u32 row = idx >> 12;                                  // / 4096
  u32 k   = idx & (D - 1);
  u32 c = k >> 8, h = (k >> 4) & 15, w = k & 15;
  const float* mu = (row < 128) ? mu_a : mu_b;
  const float* lv = (row < 128) ? lv_a : lv_b;
  u32 n = row & 127u;
  u32 base = ((n * 16u + c) * 32u + h * 2u) * 32u + w * 2u;

  float m00 = mu[base],      m01 = mu[base + 1];
  float m10 = mu[base + 32], m11 = mu[base + 33];
  float l00 = lv[base],      l01 = lv[base + 1];
  float l10 = lv[base + 32], l11 = lv[base + 33];

  X[idx] = 0.25f * (m00 + m01 + m10 + m11);
  // avgpool(exp(lv)) * (1/4)  ==  sum(exp)/16
  float e = __builtin_amdgcn_exp2f(l00 * LOG2E)
          + __builtin_amdgcn_exp2f(l01 * LOG2E)
          + __builtin_amdgcn_exp2f(l10 * LOG2E)
          + __builtin_amdgcn_exp2f(l11 * LOG2E);
  V[idx] = 0.0625f * e;
}

// ---------------------------------------------------------------------------
// TDM: load a 16 x KT f32 tile (row stride 4096 floats) from a [256][4096]
// matrix into contiguous LDS. 2-D descriptor, groups 2/3/4 zeroed.
// ---------------------------------------------------------------------------
__device__ __forceinline__ void tdm_load_tile(const float* gbase, u32 row0,
                                              u32 k0, u32 lds_byte_addr) {
  u64 gaddr = (u64)(uintptr_t)gbase + 4ull * ((u64)row0 * (u64)D + (u64)k0);
  v4u g0;
  g0.x = 1u;                                        // count=1, user descriptor
  g0.y = lds_byte_addr;                             // LDS byte address
  g0.z = (u32)(gaddr & 0xFFFFFFFFull);              // global addr [31:0]
  g0.w = (u32)((gaddr >> 32) & 0x1FFFFFFull)        // global addr [56:32]
       | (2u << 30);                                // type = 2 ("image")
  v8i g1;
  g1[0] = (int)(2u << 16);                          // data_size = 2 (4 bytes)
  g1[1] = (int)(((u32)D & 0xFFFFu) << 16);          // tensor_dim0[15:0] @ [31:16]
  g1[2] = (int)(((u32)D >> 16) | (((u32)NROWS & 0xFFFFu) << 16)); // td0 hi, td1 lo
  g1[3] = (int)(((u32)NROWS >> 16) | ((u32)KT << 16));            // td1 hi, tile_dim0
  g1[4] = (int)(16u);                               // tile_dim1=16, tile_dim2=0
  g1[5] = (int)((u32)D);                            // tensor_dim0_stride[31:0]
  g1[6] = 0;                                        // stride hi, td1_stride lo
  g1[7] = 0;
  v4i g2 = {0, 0, 0, 0};
  v4i g3 = {0, 0, 0, 0};
  v8i g4 = {0, 0, 0, 0, 0, 0, 0, 0};
  __builtin_amdgcn_tensor_load_to_lds(g0, g1, g2, g3, g4, 0);
}

// ---------------------------------------------------------------------------
// Stage 2: signed pair sum. Grid = 16 iTiles * 16 jTiles * 8 kSlabs = 2048.
// Block: 256 threads, thread t owns pair (i = it*16 + t/16, j = jt*16 + t%16).
// Per element: d = xi-xj, s = vi+vj,
//   contribution = rsqrt(s) * exp2(-0.5*log2e * d*d / s)       (2 trans ops)
// TDM double-buffered: wave 0 issues chunk c+1 DMA before computing chunk c.
// ---------------------------------------------------------------------------
__global__ __launch_bounds__(256) void pair_sum_kernel(
    const float* __restrict__ X, const float* __restrict__ V,
    float* __restrict__ partials) {
  __shared__ float sXi[2][16 * KT];
  __shared__ float sVi[2][16 * KT];
  __shared__ float sXj[2][16 * KT];
  __shared__ float sVj[2][16 * KT];
  __shared__ float red[256];

  const u32 bid = blockIdx.x;
  const u32 kb = bid % KBLK;
  const u32 jt = (bid / KBLK) & 15u;
  const u32 it = bid / (KBLK * 16u);
  const u32 t  = threadIdx.x;
  const u32 il = t >> 4, jl = t & 15u;
  const float sign = ((it < 8u) == (jt < 8u)) ? 1.0f : -1.0f;
  const u32 kbase = kb * KCHUNK;

  // Prologue: DMA chunk 0 into buffer 0 (wave 0 drives the TDM).
  if (t < 32u) {
    tdm_load_tile(X, it * 16u, kbase, (u32)(uintptr_t)&sXi[0][0]);
    tdm_load_tile(V, it * 16u, kbase, (u32)(uintptr_t)&sVi[0][0]);
    tdm_load_tile(X, jt * 16u, kbase, (u32)(uintptr_t)&sXj[0][0]);
    tdm_load_tile(V, jt * 16u, kbase, (u32)(uintptr_t)&sVj[0][0]);
    __builtin_amdgcn_s_wait_tensorcnt(0);
  }
  __syncthreads();

  float acc = 0.0f;
  for (u32 c = 0; c < NCHUNK; ++c) {
    const u32 cur = c & 1u;
    // Prefetch chunk c+1 into the other buffer while we compute chunk c.
    if ((c + 1u < NCHUNK) && (t < 32u)) {
      const u32 nxt = cur ^ 1u;
      const u32 k1 = kbase + (c + 1u) * KT;
      tdm_load_tile(X, it * 16u, k1, (u32)(uintptr_t)&sXi[nxt][0]);
      tdm_load_tile(V, it * 16u, k1, (u32)(uintptr_t)&sVi[nxt][0]);
      tdm_load_tile(X, jt * 16u, k1, (u32)(uintptr_t)&sXj[nxt][0]);
      tdm_load_tile(V, jt * 16u, k1, (u32)(uintptr_t)&sVj[nxt][0]);
    }

    const float* xi = &sXi[cur][il * KT];
    const float* vi = &sVi[cur][il * KT];
    const float* xj = &sXj[cur][jl * KT];
    const float* vj = &sVj[cur][jl * KT];
#pragma unroll 4
    for (u32 kk = 0; kk < KT; kk += 4) {
      float4 x1 = *(const float4*)(xi + kk);
      float4 v1 = *(const float4*)(vi + kk);
      float4 x2 = *(const float4*)(xj + kk);
      float4 v2 = *(const float4*)(vj + kk);
#define PAIR_TERM(a, b, c_, d_)                                     \
      {                                                             \
        float dd = (a) - (b);                                       \
        float ss = (c_) + (d_);                                     \
        float qq = __builtin_amdgcn_rsqf(ss);   /* 1/sqrt(s) */     \
        float rr = qq * qq;                     /* 1/s       */     \
        acc += qq * __builtin_amdgcn_exp2f(dd * dd * rr * NEG_HALF_LOG2E); \
      }
      PAIR_TERM(x1.x, x2.x, v1.x, v2.x)
      PAIR_TERM(x1.y, x2.y, v1.y, v2.y)
      PAIR_TERM(x1.z, x2.z, v1.z, v2.z)
      PAIR_TERM(x1.w, x2.w, v1.w, v2.w)
#undef PAIR_TERM
    }

    __syncthreads();                 // all waves done reading buffer `cur`
    if (c + 1u < NCHUNK) {
      if (t < 32u) __builtin_amdgcn_s_wait_tensorcnt(0);  // next buffer landed
      __syncthreads();               // make it visible to all waves
    }
  }

  // deterministic intra-block tree reduction
  red[t] = acc;
  __syncthreads();
  for (u32 s = 128; s > 0; s >>= 1) {
    if (t < s) red[t] += red[t + s];
    __syncthreads();
  }
  if (t == 0) partials[bid] = sign * red[0];
}

// ---------------------------------------------------------------------------
// Stage 3: deterministic final reduction of 2048 block partials.
// ---------------------------------------------------------------------------
__global__ __launch_bounds__(256) void reduce_kernel(
    const float* __restrict__ partials, int n, float* __restrict__ out) {
  __shared__ float red[256];
  float a = 0.0f;
  for (int i = threadIdx.x; i < n; i += 256) a += partials[i];
  red[threadIdx.x] = a;
  __syncthreads();
  for (int s = 128; s > 0; s >>= 1) {
    if ((int)threadIdx.x < s) red[threadIdx.x] += red[threadIdx.x + s];
    __syncthreads();
  }
  if (threadIdx.x == 0) out[0] = red[0];
}

extern "C" void kernel_launch(void* const* d_in, const int* in_sizes, int n_in,
                              void* d_out, int out_size, void* d_ws, size_t ws_size,
                              hipStream_t stream) {
  const float* mu_a = (const float*)d_in[0];
  const float* lv_a = (const float*)d_in[1];
  const float* mu_b = (const float*)d_in[2];
  const float* lv_b = (const float*)d_in[3];

  float* X = (float*)d_ws;                    // [256][4096]
  float* V = X + (size_t)NROWS * D;           // [256][4096]
  float* partials = V + (size_t)NROWS * D;    // [2048]

  pool_kernel<<<(NROWS * D) / 256, 256, 0, stream>>>(mu_a, lv_a, mu_b, lv_b, X, V);
  pair_sum_kernel<<<16 * 16 * KBLK, 256, 0, stream>>>(X, V, partials);
  reduce_kernel<<<1, 256, 0, stream>>>(partials, 16 * 16 * KBLK, (float*)d_out);
}